// VectorQuantizer_12945031430946
// MI455X (gfx1250) — compile-verified
//
#include <hip/hip_runtime.h>
#include <hip/hip_bf16.h>

typedef __attribute__((ext_vector_type(2))) float v2f;
typedef __attribute__((ext_vector_type(4))) float v4f;
typedef __attribute__((ext_vector_type(8))) float v8f;

#define B_DIM 16
#define C_DIM 256
#define T_DIM 4096
#define K_CODES 1024
#define N_ROWS (B_DIM * T_DIM)            // 65536
#define OUT_ELEMS (B_DIM * C_DIM * T_DIM) // 16777216
#define CP 260                            // padded LDS row stride: 16B-aligned rows,
                                          // conflict-free frag reads (banks 4l+k / 4l+k+2)

// ---------------------------------------------------------------------------
// 1) esq[k] = ||embedding[k]||^2   (also diag of E E^T)
// ---------------------------------------------------------------------------
__global__ __launch_bounds__(256) void vq_esq_kernel(const float* __restrict__ emb,
                                                     float* __restrict__ esq) {
    int k = blockIdx.x * 256 + threadIdx.x;   // grid = 4 -> k in [0,1024)
    float s = 0.0f;
    const float* row = emb + (size_t)k * C_DIM;
    #pragma unroll 8
    for (int c = 0; c < C_DIM; ++c) { float v = row[c]; s += v * v; }
    esq[k] = s;
}

// ---------------------------------------------------------------------------
// 2) argmin over codes via V_WMMA_F32_16X16X4_F32
//    block = 64 threads (2 waves); each wave handles 16 rows of z^T.
//    A panel (16 rows x 256 k) lives in 128 VGPRs; nc-loop streams only B
//    fragments from LDS (one ds_load_b64 per wmma); e-tile staged with b128.
// ---------------------------------------------------------------------------
__global__ __launch_bounds__(64) void vq_argmin_kernel(const float* __restrict__ z,
                                                       const float* __restrict__ emb,
                                                       const float* __restrict__ esq,
                                                       int* __restrict__ idx_out,
                                                       float* __restrict__ counts) {
    __shared__ float lds_z[2 * 16 * CP];   // [wave*16 + n][c], padded
    __shared__ float lds_e[16 * CP];       // [code][c], padded

    const int tid  = threadIdx.x;
    const int wv   = tid >> 5;
    const int lane = tid & 31;
    const int hf   = lane >> 4;            // 0: lanes 0-15, 1: lanes 16-31
    const int l16  = lane & 15;

    const int row0 = blockIdx.x * 32;      // 32 rows per block; 4096 % 32 == 0 -> same batch b
    const int b    = row0 >> 12;
    const int t0   = row0 & (T_DIM - 1);

    // Stage z^T tile (32 rows x 256) into LDS with b128 global loads
    // (t-contiguous), scalar LDS stores do the transpose.
    for (int f = tid; f < (32 * C_DIM) / 4; f += 64) {
        int c  = f >> 3;                   // 0..255
        int n4 = (f & 7) << 2;             // 0,4,...,28
        v4f v = *(const v4f*)(z + ((size_t)(b * C_DIM + c) << 12) + t0 + n4);
        lds_z[(n4 + 0) * CP + c] = v.x;
        lds_z[(n4 + 1) * CP + c] = v.y;
        lds_z[(n4 + 2) * CP + c] = v.z;
        lds_z[(n4 + 3) * CP + c] = v.w;
    }
    __syncthreads();

    // Preload A fragments for the whole k range into registers.
    // A (16x4 f32): lanes 0-15 hold K={0,1}, lanes 16-31 K={2,3}, M = lane&15.
    const float* zp = &lds_z[(wv * 16 + l16) * CP + 2 * hf];
    v2f areg[64];
    #pragma unroll
    for (int kc = 0; kc < 64; ++kc)
        areg[kc] = *(const v2f*)(zp + 4 * kc);

    float best[8];
    int   bidx[8];
    #pragma unroll
    for (int j = 0; j < 8; ++j) { best[j] = 3.402823466e38f; bidx[j] = 0; }

    for (int nc = 0; nc < K_CODES / 16; ++nc) {
        __syncthreads();                   // protect lds_e reuse
        // 16 codes x 256 floats, b128 both sides (CP*4 and 16*code*CP*4 are 16B-aligned)
        for (int f = tid; f < (16 * C_DIM) / 4; f += 64) {
            int code = f >> 6;
            int c4   = (f & 63) << 2;
            *(v4f*)&lds_e[code * CP + c4] =
                *(const v4f*)(emb + ((size_t)(nc * 16 + code) << 8) + c4);
        }
        __syncthreads();

        // B fragment (4x16 f32): lane holds column n = lane&15, K split like A
        const float* ep = &lds_e[l16 * CP + 2 * hf];

        v8f acc = {};                      // 16x16 f32 accumulator tile
        #pragma unroll
        for (int kc = 0; kc < 64; ++kc) {
            v2f bb = *(const v2f*)(ep + 4 * kc);
            acc = __builtin_amdgcn_wmma_f32_16x16x4_f32(
                false, areg[kc], false, bb, (short)0, acc, false, false);
        }

        // dist = ||e||^2 - 2 z.e  (||z||^2 is constant per row -> irrelevant for argmin)
        const int col = nc * 16 + l16;
        const float es = esq[col];
        #pragma unroll
        for (int j = 0; j < 8; ++j) {
            float dist = es - 2.0f * acc[j];   // acc[j]: row j + 8*hf, col lane&15
            if (dist < best[j]) { best[j] = dist; bidx[j] = col; }
        }
    }

    // Min-reduce across the 16 lanes of each half (cols), first-index tie-break.
    #pragma unroll
    for (int j = 0; j < 8; ++j) {
        #pragma unroll
        for (int m = 8; m >= 1; m >>= 1) {
            float ov = __shfl_xor(best[j], m, 32);
            int   oi = __shfl_xor(bidx[j], m, 32);
            if (ov < best[j] || (ov == best[j] && oi < bidx[j])) { best[j] = ov; bidx[j] = oi; }
        }
        if (l16 == 0) {
            int row = row0 + wv * 16 + hf * 8 + j;
            idx_out[row] = bidx[j];
            atomicAdd(&counts[bidx[j]], 1.0f);
        }
    }
}

// ---------------------------------------------------------------------------
// 3) gather codebook rows (straight-through) + summed squared error
// ---------------------------------------------------------------------------
__global__ __launch_bounds__(256) void vq_gather_loss_kernel(const float* __restrict__ z,
                                                             const float* __restrict__ emb,
                                                             const int* __restrict__ idx,
                                                             float* __restrict__ out,
                                                             float* __restrict__ loss_acc) {
    size_t i = (size_t)blockIdx.x * 256 + threadIdx.x;   // < OUT_ELEMS, coalesced in z layout
    int t = (int)(i & (T_DIM - 1));
    int c = (int)((i >> 12) & (C_DIM - 1));
    int b = (int)(i >> 20);
    int n = (b << 12) | t;

    int code = idx[n];
    float e  = emb[(size_t)code * C_DIM + c];
    float zv = z[i];
    out[i] = zv + (e - zv);                // matches z + sg(vq - z) bitwise
    float d = e - zv;
    d = d * d;

    __shared__ float red[8];
    #pragma unroll
    for (int m = 16; m >= 1; m >>= 1) d += __shfl_xor(d, m, 32);
    if ((threadIdx.x & 31) == 0) red[threadIdx.x >> 5] = d;
    __syncthreads();
    if (threadIdx.x < 8) {
        float v = red[threadIdx.x];
        #pragma unroll
        for (int m = 4; m >= 1; m >>= 1) v += __shfl_xor(v, m, 32);
        if (threadIdx.x == 0) atomicAdd(loss_acc, v);
    }
}

// ---------------------------------------------------------------------------
// 4) sparsity: sum over rows of (logsumexp(E E^T row) - diag); one wave per row
// ---------------------------------------------------------------------------
__global__ __launch_bounds__(256) void vq_sparsity_kernel(const float* __restrict__ emb,
                                                          const float* __restrict__ esq,
                                                          float* __restrict__ spars_acc) {
    __shared__ float er[8][C_DIM];
    const int wv   = threadIdx.x >> 5;
    const int lane = threadIdx.x & 31;
    const int r    = blockIdx.x * 8 + wv;            // grid = 128

    for (int c = lane; c < C_DIM; c += 32) er[wv][c] = emb[(size_t)r * C_DIM + c];
    __syncthreads();

    float mx = -3.402823466e38f, sum = 0.0f;
    for (int kk = 0; kk < K_CODES / 32; ++kk) {
        int k = kk * 32 + lane;
        const float* ek = emb + (size_t)k * C_DIM;
        float acc = 0.0f;
        #pragma unroll 8
        for (int c = 0; c < C_DIM; ++c) acc += er[wv][c] * ek[c];
        if (acc > mx) { sum = sum * __expf(mx - acc) + 1.0f; mx = acc; }
        else          { sum += __expf(acc - mx); }
    }
    #pragma unroll
    for (int m = 16; m >= 1; m >>= 1) {
        float omx = __shfl_xor(mx, m, 32);
        float osm = __shfl_xor(sum, m, 32);
        float nmx = fmaxf(mx, omx);
        sum = sum * __expf(mx - nmx) + osm * __expf(omx - nmx);
        mx = nmx;
    }
    if (lane == 0) atomicAdd(spars_acc, mx + __logf(sum) - esq[r]);
}

// ---------------------------------------------------------------------------
// 5) finalize scalars: losses, perplexity, sparsity mean
// ---------------------------------------------------------------------------
__global__ __launch_bounds__(256) void vq_finalize_kernel(const float* __restrict__ counts,
                                                          const float* __restrict__ loss_acc,
                                                          const float* __restrict__ spars_acc,
                                                          float* __restrict__ outs) {
    __shared__ float red[8];
    float s = 0.0f;
    for (int k = threadIdx.x; k < K_CODES; k += 256) {
        float p = counts[k] * (1.0f / (float)N_ROWS);
        s += p * __logf(p + 1e-10f);
    }
    #pragma unroll
    for (int m = 16; m >= 1; m >>= 1) s += __shfl_xor(s, m, 32);
    if ((threadIdx.x & 31) == 0) red[threadIdx.x >> 5] = s;
    __syncthreads();
    if (threadIdx.x == 0) {
        float tot = 0.0f;
        #pragma unroll
        for (int w = 0; w < 8; ++w) tot += red[w];
        float loss = *loss_acc;
        outs[0] = loss;                    // z_qut_loss
        outs[1] = loss;                    // z_enc_loss (numerically identical)
        outs[2] = __expf(-tot);            // perplexity
        outs[3] = *spars_acc * (1.0f / (float)K_CODES);
    }
}

// ---------------------------------------------------------------------------
extern "C" void kernel_launch(void* const* d_in, const int* in_sizes, int n_in,
                              void* d_out, int out_size, void* d_ws, size_t ws_size,
                              hipStream_t stream) {
    const float* z   = (const float*)d_in[0];   // (16, 256, 4096)
    const float* emb = (const float*)d_in[1];   // (1024, 256)
    float* out = (float*)d_out;

    // workspace layout
    char* ws = (char*)d_ws;
    float* esq       = (float*)(ws + 0);        // 1024 f
    float* counts    = (float*)(ws + 4096);     // 1024 f
    float* loss_acc  = (float*)(ws + 8192);     // 1 f
    float* spars_acc = (float*)(ws + 8196);     // 1 f
    int*   idx       = (int*)(ws + 8448);       // 65536 i32

    // zero the histogram + accumulators (capture-safe)
    hipMemsetAsync(ws + 4096, 0, 4096 + 8, stream);

    vq_esq_kernel<<<K_CODES / 256, 256, 0, stream>>>(emb, esq);
    vq_argmin_kernel<<<N_ROWS / 32, 64, 0, stream>>>(z, emb, esq, idx, counts);
    vq_gather_loss_kernel<<<OUT_ELEMS / 256, 256, 0, stream>>>(z, emb, idx, out, loss_acc);
    vq_sparsity_kernel<<<K_CODES / 8, 256, 0, stream>>>(emb, esq, spars_acc);
    vq_finalize_kernel<<<1, 256, 0, stream>>>(counts, loss_acc, spars_acc, out + OUT_ELEMS);
}